// TtMistralAttention_29961691856935
// MI455X (gfx1250) — compile-verified
//
#include <hip/hip_runtime.h>
#include <hip/hip_bf16.h>
#include <math.h>

typedef float v2f __attribute__((ext_vector_type(2)));
typedef float v8f __attribute__((ext_vector_type(8)));

#define DIM   4096
#define NH    32
#define NKV   8
#define HD    128
#define SW    4096
#define BATCH 32
#define QKV_N (DIM + 2 * NKV * HD)   // 6144

static __device__ __forceinline__ v8f wmma_f32(v2f a, v2f b, v8f c) {
  // V_WMMA_F32_16X16X4_F32 : D = A(16x4) * B(4x16) + C(16x16)
  return __builtin_amdgcn_wmma_f32_16x16x4_f32(false, a, false, b, (short)0, c,
                                               false, false);
}

static __device__ __forceinline__ float wave_max32(float v) {
#pragma unroll
  for (int m = 16; m >= 1; m >>= 1) v = fmaxf(v, __shfl_xor(v, m, 32));
  return v;
}
static __device__ __forceinline__ float wave_sum32(float v) {
#pragma unroll
  for (int m = 16; m >= 1; m >>= 1) v += __shfl_xor(v, m, 32);
  return v;
}

// ---------------------------------------------------------------------------
// C(32 x N) = A(32 x K, row-major) * W(K x N, row-major), fp32 WMMA 16x16x4.
// One wave owns 16 output columns; two accumulators cover all 32 rows so W
// streams through exactly once (W is the bandwidth term).
//   A fragments: per-lane pointers bumped +4 floats/iter -> immediate-offset
//                global_load_b64 after unrolling.
//   B fragments: one 32-bit element index against two scalar bases (W, W+N)
//                -> saddr-form global_load_b32 ... scale_offset, one
//                v_add_nc_u32 per K-step.
// ---------------------------------------------------------------------------
__global__ void gemm32_wmma(const float* __restrict__ A,
                            const float* __restrict__ W,
                            float* __restrict__ C, int K, int N) {
  const int lane = threadIdx.x & 31;
  const int wave = threadIdx.x >> 5;
  const int n0 = (blockIdx.x * (blockDim.x >> 5) + wave) * 16;
  if (n0 >= N) return;

  const int lr   = lane & 15;        // row (A) / column (B,C) within tile
  const int koff = (lane >> 4) * 2;  // lanes 16-31 carry K+2,K+3

  const float* a0p = A + (unsigned)lr * K + koff;   // A rows 0..15
  const float* a1p = a0p + 16u * (unsigned)K;       // A rows 16..31
  const float* Wn  = W + N;                         // second scalar base
  unsigned bIdx = (unsigned)koff * N + n0 + lr;     // W rows koff, koff+1

  v8f acc0 = {};
  v8f acc1 = {};

  const unsigned bStep = 4u * (unsigned)N;
#pragma unroll 8
  for (int k = 0; k < K; k += 4) {
    v2f a0 = *(const v2f*)a0p;
    v2f a1 = *(const v2f*)a1p;
    v2f b;
    b.x = W[bIdx];
    b.y = Wn[bIdx];
    acc0 = wmma_f32(a0, b, acc0);
    acc1 = wmma_f32(a1, b, acc1);
    a0p += 4;
    a1p += 4;
    bIdx += bStep;
  }

  const int half8 = (lane >> 4) * 8;  // lanes 16-31 hold rows M+8
  unsigned cIdx = (unsigned)half8 * N + n0 + lr;
#pragma unroll
  for (int r = 0; r < 8; ++r) {
    C[cIdx + (unsigned)r * N]        = acc0[r];
    C[cIdx + (unsigned)(r + 16) * N] = acc1[r];
  }
}

// ---------------------------------------------------------------------------
// RoPE on q/k, plain copy of v.  xqkv row layout: [q 4096 | k 1024 | v 1024].
// grid (48, 32): x = head slot (0-31 q, 32-39 k, 40-47 v), y = batch token.
// ---------------------------------------------------------------------------
__global__ void rope_kernel(const float* __restrict__ xqkv,
                            const float* __restrict__ cos_c,
                            const float* __restrict__ sin_c,
                            const int* __restrict__ sp_ptr,
                            float* __restrict__ q_out,
                            float* __restrict__ k_new,
                            float* __restrict__ v_new) {
  const int b = blockIdx.y;
  const int h = blockIdx.x;
  const int d = threadIdx.x;  // 0..127
  const int sp = sp_ptr[0];
  const float c = cos_c[(size_t)sp * HD + d];
  const float s = sin_c[(size_t)sp * HD + d];
  const float* row = xqkv + (size_t)b * QKV_N;

  if (h < NH) {  // query heads
    const float* u = row + h * HD;
    float rot = (d < HD / 2) ? -u[d + HD / 2] : u[d - HD / 2];
    q_out[((size_t)b * NH + h) * HD + d] = u[d] * c + rot * s;
  } else if (h < NH + NKV) {  // key heads
    const int kvh = h - NH;
    const float* u = row + DIM + kvh * HD;
    float rot = (d < HD / 2) ? -u[d + HD / 2] : u[d - HD / 2];
    k_new[((size_t)b * NKV + kvh) * HD + d] = u[d] * c + rot * s;
  } else {  // value heads: copy
    const int kvh = h - NH - NKV;
    v_new[((size_t)b * NKV + kvh) * HD + d] =
        row[DIM + NKV * HD + kvh * HD + d];
  }
}

// ---------------------------------------------------------------------------
// Flash-style GQA decode attention.  grid (NKV, BATCH), block 256 (8 waves).
// Each wave walks 16-wide L chunks with a private online softmax; scores via
// fp32 WMMA (4 real query rows padded to 16), P*V via coalesced FMA; wave
// partials merged through LDS at the end.  The freshly RoPE'd K/V row is
// substituted at current_pos via a base-pointer select (inputs must not be
// mutated in place).
// ---------------------------------------------------------------------------
__global__ void attn_kernel(const float* __restrict__ q_rot,
                            const float* __restrict__ k_new,
                            const float* __restrict__ v_new,
                            const float* __restrict__ cache_k,
                            const float* __restrict__ cache_v,
                            const float* __restrict__ mask,
                            const int* __restrict__ sp_ptr,
                            const int* __restrict__ cp_ptr,
                            float* __restrict__ attn_out) {
  __shared__ float q_s[4 * HD];       // 4 query rows for this kv head
  __shared__ float o_s[8 * 4 * HD];   // per-wave partial outputs
  __shared__ float m_s[8 * 4];
  __shared__ float l_s[8 * 4];

  const int kv  = blockIdx.x;
  const int b   = blockIdx.y;
  const int tid = threadIdx.x;
  const int lane = tid & 31;
  const int wave = tid >> 5;
  const int lr   = lane & 15;
  const int half = lane >> 4;
  const int koff = half * 2;

  const int sp  = sp_ptr[0];
  const int cur = cp_ptr[0];
  int padded = ((sp + 1 + 31) / 32) * 32;
  if (padded > SW) padded = SW;

  // stage the 4 grouped query rows into LDS
#pragma unroll
  for (int i = tid; i < 4 * HD; i += 256)
    q_s[i] = q_rot[((size_t)b * NH + kv * 4) * HD + i];
  __syncthreads();

  const float* kslice = cache_k + (((size_t)b * NKV + kv) * SW) * HD;
  const float* vslice = cache_v + (((size_t)b * NKV + kv) * SW) * HD;
  const float* kn = k_new + ((size_t)b * NKV + kv) * HD;
  const float* vn = v_new + ((size_t)b * NKV + kv) * HD;
  const float* mbase = mask + ((size_t)b * NH + kv * 4) * padded;
  const float scale = 0.08838834764831845f;  // 1/sqrt(128)

  float m_r[4] = {-1e30f, -1e30f, -1e30f, -1e30f};
  float l_r[4] = {0.f, 0.f, 0.f, 0.f};
  float o_acc[4][4] = {};

  for (int l0 = wave * 16; l0 < padded; l0 += 128) {
    // ---- scores S(16x16) = Qpad(16x128) * K^T, WMMA over K-dim ----
    const int lrow = l0 + lr;
    const float* krow = (lrow == cur) ? kn : (kslice + (unsigned)lrow * HD);

    v8f s0 = {};
    v8f s1 = {};
#pragma unroll 8
    for (int kk = 0; kk < HD; kk += 8) {
      v2f a0 = (lr < 4) ? *(const v2f*)(q_s + lr * HD + kk + koff) : (v2f){0.f, 0.f};
      v2f b0 = *(const v2f*)(krow + kk + koff);
      s0 = wmma_f32(a0, b0, s0);
      v2f a1 = (lr < 4) ? *(const v2f*)(q_s + lr * HD + kk + 4 + koff) : (v2f){0.f, 0.f};
      v2f b1 = *(const v2f*)(krow + kk + 4 + koff);
      s1 = wmma_f32(a1, b1, s1);
    }
    v8f s = s0 + s1;  // rows 0..3 valid (lanes 0..15), rest zero

    // ---- online softmax update (stats replicated to all 32 lanes) ----
    float pr[4];
#pragma unroll
    for (int r = 0; r < 4; ++r) {
      float sv;
      if (half == 0)
        sv = s[r] * scale + mbase[(unsigned)r * padded + l0 + lr];
      else
        sv = -1e30f;
      float mx   = wave_max32(sv);
      float mnew = fmaxf(m_r[r], mx);
      float alpha = __expf(m_r[r] - mnew);
      float p = (half == 0) ? __expf(sv - mnew) : 0.f;
      float ps = wave_sum32(p);
      l_r[r] = l_r[r] * alpha + ps;
      m_r[r] = mnew;
      pr[r]  = p;
#pragma unroll
      for (int t = 0; t < 4; ++t) o_acc[r][t] *= alpha;
    }

    // ---- O += P(4x16) * V(16x128), coalesced V rows + shfl broadcast ----
#pragma unroll 4
    for (int j = 0; j < 16; ++j) {
      const int row = l0 + j;
      const float* vrow = (row == cur) ? vn : (vslice + (unsigned)row * HD);
      float vv[4];
#pragma unroll
      for (int t = 0; t < 4; ++t) vv[t] = vrow[t * 32 + lane];
#pragma unroll
      for (int r = 0; r < 4; ++r) {
        float pj = __shfl(pr[r], j, 32);
#pragma unroll
        for (int t = 0; t < 4; ++t) o_acc[r][t] = fmaf(pj, vv[t], o_acc[r][t]);
      }
    }
  }

  // ---- publish per-wave partials, merge across the 8 waves ----
#pragma unroll
  for (int r = 0; r < 4; ++r)
#pragma unroll
    for (int t = 0; t < 4; ++t)
      o_s[(wave * 4 + r) * HD + t * 32 + lane] = o_acc[r][t];
  if (lane == 0) {
#pragma unroll
    for (int r = 0; r < 4; ++r) {
      m_s[wave * 4 + r] = m_r[r];
      l_s[wave * 4 + r] = l_r[r];
    }
  }
  __syncthreads();

  for (int idx = tid; idx < 4 * HD; idx += 256) {
    const int r = idx >> 7;
    const int d = idx & (HD - 1);
    float M = -1e30f;
#pragma unroll
    for (int w = 0; w < 8; ++w) M = fmaxf(M, m_s[w * 4 + r]);
    float L = 0.f, O = 0.f;
#pragma unroll
    for (int w = 0; w < 8; ++w) {
      float f = __expf(m_s[w * 4 + r] - M);
      L += l_s[w * 4 + r] * f;
      O += o_s[(w * 4 + r) * HD + d] * f;
    }
    attn_out[((size_t)b * NH + kv * 4 + r) * HD + d] = O / L;
  }
}

// ---------------------------------------------------------------------------
extern "C" void kernel_launch(void* const* d_in, const int* in_sizes, int n_in,
                              void* d_out, int out_size, void* d_ws, size_t ws_size,
                              hipStream_t stream) {
  const float* x        = (const float*)d_in[0];  // (32, 4096)
  const float* wqkv     = (const float*)d_in[1];  // (4096, 6144)
  const float* wo       = (const float*)d_in[2];  // (4096, 4096)
  const float* cache_k  = (const float*)d_in[3];  // (32, 8, 4096, 128)
  const float* cache_v  = (const float*)d_in[4];
  const float* cos_c    = (const float*)d_in[5];  // (8192, 128)
  const float* sin_c    = (const float*)d_in[6];
  const float* amask    = (const float*)d_in[7];  // (32, 32, 1, padded)
  const int*   sp       = (const int*)d_in[8];
  const int*   cp       = (const int*)d_in[9];

  float* ws     = (float*)d_ws;
  float* xqkv   = ws;                                // 32*6144  = 196608
  float* qrot   = xqkv + (size_t)BATCH * QKV_N;      // 32*32*128 = 131072
  float* kn     = qrot + (size_t)BATCH * NH * HD;    // 32*8*128  = 32768
  float* vn     = kn + (size_t)BATCH * NKV * HD;     // 32768
  float* attn_o = vn + (size_t)BATCH * NKV * HD;     // 32*4096   = 131072

  // 1) xqkv = x @ wqkv   (384 column tiles -> 48 blocks x 8 waves)
  gemm32_wmma<<<QKV_N / (16 * 8), 256, 0, stream>>>(x, wqkv, xqkv, DIM, QKV_N);

  // 2) RoPE + split q/k/v
  rope_kernel<<<dim3(NH + 2 * NKV, BATCH), HD, 0, stream>>>(
      xqkv, cos_c, sin_c, sp, qrot, kn, vn);

  // 3) flash attention over the fp32 cache
  attn_kernel<<<dim3(NKV, BATCH), 256, 0, stream>>>(
      qrot, kn, vn, cache_k, cache_v, amask, sp, cp, attn_o);

  // 4) d_out = attn_o @ wo
  gemm32_wmma<<<DIM / (16 * 8), 256, 0, stream>>>(attn_o, wo, (float*)d_out,
                                                  DIM, DIM);
}